// NAFLeafNetwork_9680856285709
// MI455X (gfx1250) — compile-verified
//
#include <hip/hip_runtime.h>
#include <hip/hip_bf16.h>
#include <cstdint>

typedef __attribute__((ext_vector_type(2))) float v2f;
typedef __attribute__((ext_vector_type(8))) float v8f;

#define S_DIM 512
#define B_DIM 2048
#define T_DIM 32
#define F_DIM 64
#define H_DIM 128
#define NEG_INF (-1e20f)

#if defined(__gfx1250__) && __has_builtin(__builtin_amdgcn_global_load_async_to_lds_b128)
#define HAVE_ASYNC_LDS 1
// Exact parameter types per hipcc diagnostic:
//   arg0: int __attribute__((vector_size(16))) addrspace(1)*   (global)
//   arg1: same vector type in addrspace(3)                      (LDS)
typedef int v4i_gcc __attribute__((vector_size(16)));
typedef __attribute__((address_space(1))) v4i_gcc* gas_v4i;
typedef __attribute__((address_space(3))) v4i_gcc* las_v4i;
#else
#define HAVE_ASYNC_LDS 0
#endif

// D = A(16x4) * B(4x16) + C, all f32, full precision on the matrix unit.
__device__ __forceinline__ v8f wmma_k4(v2f a, v2f b, v8f c) {
    return __builtin_amdgcn_wmma_f32_16x16x4_f32(
        /*neg_a=*/false, a, /*neg_b=*/false, b,
        /*c_mod=*/(short)0, c, /*reuse_a=*/false, /*reuse_b=*/false);
}

// Overflow-safe tanh: 1 - 2/(e^{2x}+1)  ->  +/-1 at the extremes.
__device__ __forceinline__ float fast_tanh(float x) {
    float e = __expf(2.0f * x);
    return 1.0f - 2.0f / (e + 1.0f);
}

// ---------------------------------------------------------------------------
// Kernel 1: encode 2560 rows ([X; background_X]) through Linear-Tanh-Linear.
// 4 waves/block, each wave owns 16 rows. Both GEMMs via V_WMMA_F32_16X16X4_F32.
// ---------------------------------------------------------------------------
__global__ __launch_bounds__(128) void encode_kernel(
    const float* __restrict__ X,    // 512 x 64
    const float* __restrict__ bgX,  // 2048 x 64
    const float* __restrict__ W1,   // 64 x 128
    const float* __restrict__ b1,   // 128
    const float* __restrict__ W2,   // 128 x 128
    const float* __restrict__ b2,   // 128
    float* __restrict__ enc)        // 2560 x 128
{
    const int wave  = threadIdx.x >> 5;
    const int lane  = threadIdx.x & 31;
    const int l15   = lane & 15;
    const int khalf = lane >> 4;           // which K-pair this half-wave holds

    const int r0 = (blockIdx.x * 4 + wave) * 16;   // first of 16 rows (16 | 512)
    const float* src;
    int srow;
    if (r0 < S_DIM) { src = X;   srow = r0; }
    else            { src = bgX; srow = r0 - S_DIM; }

    __shared__ float lds_h[4][16][H_DIM + 4];      // per-wave hidden staging
    float (*hrow)[H_DIM + 4] = lds_h[wave];

    // ---- GEMM1: (16 x 64) @ (64 x 128), 8 N-tiles, K-steps of 4 ----
    v8f acc[8];
#pragma unroll
    for (int c = 0; c < 8; ++c) acc[c] = (v8f){0,0,0,0,0,0,0,0};

    const float* arow = src + (size_t)(srow + l15) * F_DIM;
    for (int k = 0; k < F_DIM; k += 4) {
        v2f a = *(const v2f*)(arow + k + 2 * khalf);     // A[l15][k+2h..k+2h+1]
#pragma unroll
        for (int c = 0; c < 8; ++c) {
            int col = c * 16 + l15;
            v2f b;
            b.x = W1[(size_t)(k + 2 * khalf)     * H_DIM + col];
            b.y = W1[(size_t)(k + 2 * khalf + 1) * H_DIM + col];
            acc[c] = wmma_k4(a, b, acc[c]);
        }
    }

    // bias + tanh, stage hidden (row-major) into LDS
#pragma unroll
    for (int c = 0; c < 8; ++c) {
        int col = c * 16 + l15;
        float bias = b1[col];
#pragma unroll
        for (int j = 0; j < 8; ++j) {
            int row = j + 8 * khalf;                     // C/D layout: VGPR j -> rows j / j+8
            hrow[row][col] = fast_tanh(acc[c][j] + bias);
        }
    }
    __syncthreads();

    // ---- GEMM2: (16 x 128) @ (128 x 128) ----
    v8f acc2[8];
#pragma unroll
    for (int c = 0; c < 8; ++c) acc2[c] = (v8f){0,0,0,0,0,0,0,0};

    for (int k = 0; k < H_DIM; k += 4) {
        v2f a;
        a.x = hrow[l15][k + 2 * khalf];
        a.y = hrow[l15][k + 2 * khalf + 1];
#pragma unroll
        for (int c = 0; c < 8; ++c) {
            int col = c * 16 + l15;
            v2f b;
            b.x = W2[(size_t)(k + 2 * khalf)     * H_DIM + col];
            b.y = W2[(size_t)(k + 2 * khalf + 1) * H_DIM + col];
            acc2[c] = wmma_k4(a, b, acc2[c]);
        }
    }

#pragma unroll
    for (int c = 0; c < 8; ++c) {
        int col = c * 16 + l15;
        float bias = b2[col];
#pragma unroll
        for (int j = 0; j < 8; ++j) {
            int row = j + 8 * khalf;
            enc[(size_t)(r0 + row) * H_DIM + col] = acc2[c][j] + bias;
        }
    }
}

// ---------------------------------------------------------------------------
// Kernel 2: dots(512 x 2048) = X_enc @ bg_enc^T, K=128.
// Block = 8 waves covering M=32 x N=256; each wave: one 16x64 strip (4 tiles).
// ---------------------------------------------------------------------------
__global__ __launch_bounds__(256) void dots_kernel(
    const float* __restrict__ enc,   // [X_enc(512x128); bg_enc(2048x128)]
    float* __restrict__ dots)        // 512 x 2048
{
    const int wave  = threadIdx.x >> 5;
    const int lane  = threadIdx.x & 31;
    const int l15   = lane & 15;
    const int khalf = lane >> 4;

    const int m0 = blockIdx.x * 32 + (wave & 1) * 16;
    const int n0 = blockIdx.y * 256 + (wave >> 1) * 64;

    const float* xenc  = enc;
    const float* bgenc = enc + (size_t)S_DIM * H_DIM;

    v8f acc[4];
#pragma unroll
    for (int c = 0; c < 4; ++c) acc[c] = (v8f){0,0,0,0,0,0,0,0};

    const float* arow = xenc + (size_t)(m0 + l15) * H_DIM;
    for (int k = 0; k < H_DIM; k += 4) {
        v2f a = *(const v2f*)(arow + k + 2 * khalf);
#pragma unroll
        for (int c = 0; c < 4; ++c) {
            int col = n0 + c * 16 + l15;
            // B[k][n] = bg_enc[n][k]: contiguous along k -> float2 load
            v2f b = *(const v2f*)(bgenc + (size_t)col * H_DIM + k + 2 * khalf);
            acc[c] = wmma_k4(a, b, acc[c]);
        }
    }

#pragma unroll
    for (int c = 0; c < 4; ++c) {
        int col = n0 + c * 16 + l15;
#pragma unroll
        for (int j = 0; j < 8; ++j) {
            int row = m0 + j + 8 * khalf;
            dots[(size_t)row * B_DIM + col] = acc[c][j];
        }
    }
}

// ---------------------------------------------------------------------------
// Kernel 3: fused per-sample masked softmax + alpha write + leaf einsums.
// 1 block (256 thr / 8 waves) per sample. leaf_xs via WMMA over 64-wide b chunks.
// ---------------------------------------------------------------------------
__global__ __launch_bounds__(256) void leaf_kernel(
    const float* __restrict__ dots,          // 512 x 2048
    const unsigned char* __restrict__ hot,   // 512 x 2048 x 32 (bool bytes)
    const float* __restrict__ bgX,           // 2048 x 64
    const float* __restrict__ bgy,           // 2048 x 1
    float* __restrict__ leaf_xs,             // 512 x 32 x 64
    float* __restrict__ leaf_y,              // 512 x 32
    float* __restrict__ alphas)              // 512 x 2048 x 32
{
    const int s    = blockIdx.x;
    const int wave = threadIdx.x >> 5;
    const int lane = threadIdx.x & 31;
    const int l15  = lane & 15;
    const int khalf = lane >> 4;

    __shared__ float sdots[B_DIM];
    __shared__ float sm[T_DIM];
    __shared__ float sinvZ[T_DIM];
    __shared__ float astage[T_DIM][68];      // (t, b-chunk) transposed alpha tile

    const unsigned char* hs = hot + (size_t)s * B_DIM * T_DIM;

    // Phase 1: dots row -> LDS.
    // CDNA5 async global->LDS path: no VGPR round trip, tracked by ASYNCcnt.
#if HAVE_ASYNC_LDS
    {
        const float* g = dots + (size_t)s * B_DIM;
        for (int i = threadIdx.x * 4; i < B_DIM; i += 256 * 4) {
            __builtin_amdgcn_global_load_async_to_lds_b128(
                (gas_v4i)(g + i), (las_v4i)(&sdots[i]),
                /*offset=*/0, /*cpol=*/0);
        }
#if __has_builtin(__builtin_amdgcn_s_wait_asynccnt)
        __builtin_amdgcn_s_wait_asynccnt(0);
#else
        asm volatile("s_wait_asynccnt 0" ::: "memory");
#endif
    }
#else
    for (int i = threadIdx.x; i < B_DIM; i += 256)
        sdots[i] = dots[(size_t)s * B_DIM + i];
#endif
    __syncthreads();

    // Phase 2: per-tree max / Z / leaf_y (each wave owns 4 trees)
    for (int tt = 0; tt < 4; ++tt) {
        int t = wave * 4 + tt;
        float m = NEG_INF;
        for (int b = lane; b < B_DIM; b += 32) {
            float sc = hs[b * T_DIM + t] ? sdots[b] : NEG_INF;
            m = fmaxf(m, sc);
        }
#pragma unroll
        for (int off = 16; off > 0; off >>= 1)
            m = fmaxf(m, __shfl_xor(m, off, 32));

        float z = 0.f, zy = 0.f;
        for (int b = lane; b < B_DIM; b += 32) {
            float sc = hs[b * T_DIM + t] ? sdots[b] : NEG_INF;
            float e  = __expf(sc - m);
            z  += e;
            zy += e * bgy[b];
        }
#pragma unroll
        for (int off = 16; off > 0; off >>= 1) {
            z  += __shfl_xor(z,  off, 32);
            zy += __shfl_xor(zy, off, 32);
        }
        if (lane == 0) {
            sm[t]    = m;
            sinvZ[t] = 1.0f / z;
            leaf_y[(size_t)s * T_DIM + t] = zy / z;
        }
    }
    __syncthreads();

    // Warm the cache for phase 4's B fragments (global_prefetch_b8).
    __builtin_prefetch(bgX + (size_t)threadIdx.x * 64, 0, 1);

    // Phase 3: coalesced alpha writes (t fastest == memory layout)
    float* arow = alphas + (size_t)s * B_DIM * T_DIM;
    for (int i = threadIdx.x; i < B_DIM * T_DIM; i += 256) {
        int b = i >> 5, t = i & 31;
        float sc = hs[i] ? sdots[b] : NEG_INF;
        arow[i] = __expf(sc - sm[t]) * sinvZ[t];
    }

    // Phase 4: leaf_xs[t, f] = sum_b alpha[t][b] * bgX[b][f] via WMMA.
    // 8 waves cover the 32x64 output: wave -> (t-tile, f-tile).
    const int t0 = (wave & 1) * 16;
    const int f0 = (wave >> 1) * 16;
    v8f acc = (v8f){0,0,0,0,0,0,0,0};

    for (int bc = 0; bc < B_DIM; bc += 64) {
        __syncthreads();   // previous chunk fully consumed
        for (int i = threadIdx.x; i < T_DIM * 64; i += 256) {
            int bb = i >> 5, t = i & 31;
            int b  = bc + bb;
            float sc = hs[b * T_DIM + t] ? sdots[b] : NEG_INF;
            astage[t][bb] = __expf(sc - sm[t]) * sinvZ[t];
        }
        __syncthreads();

#pragma unroll
        for (int k = 0; k < 64; k += 4) {
            v2f a;
            a.x = astage[t0 + l15][k + 2 * khalf];
            a.y = astage[t0 + l15][k + 2 * khalf + 1];
            int brow = bc + k + 2 * khalf;
            v2f b;
            b.x = bgX[(size_t)brow       * F_DIM + f0 + l15];
            b.y = bgX[(size_t)(brow + 1) * F_DIM + f0 + l15];
            acc = wmma_k4(a, b, acc);
        }
    }

#pragma unroll
    for (int j = 0; j < 8; ++j) {
        int t = t0 + j + 8 * khalf;
        leaf_xs[(size_t)s * T_DIM * F_DIM + (size_t)t * F_DIM + f0 + l15] = acc[j];
    }
}

// ---------------------------------------------------------------------------
extern "C" void kernel_launch(void* const* d_in, const int* in_sizes, int n_in,
                              void* d_out, int out_size, void* d_ws, size_t ws_size,
                              hipStream_t stream) {
    (void)in_sizes; (void)n_in; (void)out_size; (void)ws_size;

    const float* X   = (const float*)d_in[0];
    const float* bgX = (const float*)d_in[1];
    const float* bgy = (const float*)d_in[2];
    const unsigned char* hot = (const unsigned char*)d_in[3];
    const float* W1  = (const float*)d_in[4];
    const float* b1  = (const float*)d_in[5];
    const float* W2  = (const float*)d_in[6];
    const float* b2  = (const float*)d_in[7];

    float* enc  = (float*)d_ws;                              // 2560 x 128
    float* dots = enc + (size_t)(S_DIM + B_DIM) * H_DIM;     // 512 x 2048

    float* out      = (float*)d_out;
    float* leaf_xs  = out;                                   // 512*32*64
    float* leaf_yp  = out + (size_t)S_DIM * T_DIM * F_DIM;   // 512*32
    float* alphas   = leaf_yp + (size_t)S_DIM * T_DIM;       // 512*2048*32

    encode_kernel<<<40, 128, 0, stream>>>(X, bgX, W1, b1, W2, b2, enc);

    dim3 g2(S_DIM / 32, B_DIM / 256);
    dots_kernel<<<g2, 256, 0, stream>>>(enc, dots);

    leaf_kernel<<<S_DIM, 256, 0, stream>>>(dots, hot, bgX, bgy,
                                           leaf_xs, leaf_yp, alphas);
}